// SparseMatmul3D_36155034698289
// MI455X (gfx1250) — compile-verified
//
#include <hip/hip_runtime.h>

// SparseMatmul3D forward: out[b,n,m] = sum_d x[b,n,d] * y[b,m,d]
// B=4, N=M=4096, D=64, fp32 in / fp32 out.
//
// Memory-bound on the 268 MB output stream -> fp32 WMMA (V_WMMA_F32_16X16X4_F32),
// 64x64 output tile per wave, non-temporal output stores.

typedef __attribute__((ext_vector_type(2))) float v2f;
typedef __attribute__((ext_vector_type(8))) float v8f;

#define BB 4
#define NN 4096
#define MM 4096
#define DD 64

// Block: 256 threads = 8 wave32s, arranged 2 (n) x 4 (m).
// Wave tile: 64 (n) x 64 (m). Macro tile: 128 (n) x 256 (m).
__global__ __launch_bounds__(256) void sparse_matmul3d_wmma_f32(
    const float* __restrict__ x,   // [B, N, D]
    const float* __restrict__ y,   // [B, M, D]
    float* __restrict__ out)       // [B, N, M]
{
    const int lane  = threadIdx.x & 31;
    const int wave  = threadIdx.x >> 5;
    const int waveN = wave >> 2;          // 0..1
    const int waveM = wave & 3;           // 0..3

    const int b  = blockIdx.z;
    const int n0 = blockIdx.y * 128 + waveN * 64;
    const int m0 = blockIdx.x * 256 + waveM * 64;

    const int half = lane >> 4;           // 0: K=k..k+1, 1: K=k+2..k+3
    const int lrow = lane & 15;           // row (A) / col (B) within 16-tile

    const float* xb = x + (size_t)b * NN * DD;
    const float* yb = y + (size_t)b * MM * DD;

    // Per-lane fragment base pointers (row-major [*,64], contiguous float2 per k-step).
    const float* ap[4];
    const float* bp[4];
#pragma unroll
    for (int i = 0; i < 4; ++i) {
        ap[i] = xb + (size_t)(n0 + i * 16 + lrow) * DD + half * 2;
        bp[i] = yb + (size_t)(m0 + i * 16 + lrow) * DD + half * 2;
    }

    v8f acc[4][4];
#pragma unroll
    for (int i = 0; i < 4; ++i)
#pragma unroll
        for (int j = 0; j < 4; ++j)
            acc[i][j] = (v8f){0.f, 0.f, 0.f, 0.f, 0.f, 0.f, 0.f, 0.f};

    // K loop: D=64, 4 per WMMA -> 16 steps, 16 WMMAs per step.
#pragma unroll 4
    for (int k = 0; k < DD; k += 4) {
        v2f af[4], bf[4];
#pragma unroll
        for (int i = 0; i < 4; ++i) {
            af[i] = *(const v2f*)(ap[i] + k);
            bf[i] = *(const v2f*)(bp[i] + k);
        }
#pragma unroll
        for (int i = 0; i < 4; ++i)
#pragma unroll
            for (int j = 0; j < 4; ++j)
                acc[i][j] = __builtin_amdgcn_wmma_f32_16x16x4_f32(
                    /*neg_a=*/false, af[i],
                    /*neg_b=*/false, bf[j],
                    /*c_mod=*/(short)0, acc[i][j],
                    /*reuse_a=*/false, /*reuse_b=*/false);
    }

    // Store: C/D layout -> VGPR r: row = r + 8*half, col = lrow (per 16x16 tile).
    // Output stream is written once and never re-read -> non-temporal stores.
#pragma unroll
    for (int i = 0; i < 4; ++i) {
        const int nbase = n0 + i * 16 + half * 8;
#pragma unroll
        for (int j = 0; j < 4; ++j) {
            const int m = m0 + j * 16 + lrow;
#pragma unroll
            for (int r = 0; r < 8; ++r) {
                const size_t off = ((size_t)b * NN + (size_t)(nbase + r)) * MM + m;
                __builtin_nontemporal_store(acc[i][j][r], out + off);
            }
        }
    }
}

extern "C" void kernel_launch(void* const* d_in, const int* in_sizes, int n_in,
                              void* d_out, int out_size, void* d_ws, size_t ws_size,
                              hipStream_t stream) {
    (void)in_sizes; (void)n_in; (void)out_size; (void)d_ws; (void)ws_size;
    const float* x = (const float*)d_in[0];
    const float* y = (const float*)d_in[1];
    float* out = (float*)d_out;

    dim3 grid(MM / 256, NN / 128, BB);   // (16, 32, 4)
    dim3 block(256);
    hipLaunchKernelGGL(sparse_matmul3d_wmma_f32, grid, block, 0, stream, x, y, out);
}